// GAT_GraphSAGE_53446573031630
// MI455X (gfx1250) — compile-verified
//
#include <hip/hip_runtime.h>

#define F 35
#define FP 48   // feature dimension padded to 48 (3 x 16 WMMA column tiles)

typedef float v2f __attribute__((ext_vector_type(2)));
typedef float v8f __attribute__((ext_vector_type(8)));

// ---- DPP16 helpers: butterfly reduce+broadcast within each 16-lane row ----
#define DPP_QUAD_XOR1   0xB1   // quad_perm [1,0,3,2]
#define DPP_QUAD_XOR2   0x4E   // quad_perm [2,3,0,1]
#define DPP_HALF_MIRROR 0x141  // lane i <-> i^7 within 8
#define DPP_ROW_MIRROR  0x140  // lane i <-> i^15 within 16

template <int CTRL>
__device__ __forceinline__ float dppmov(float x) {
  int r = __builtin_amdgcn_update_dpp(0, __float_as_int(x), CTRL, 0xF, 0xF, true);
  return __int_as_float(r);
}
__device__ __forceinline__ float row16_max(float x) {
  x = fmaxf(x, dppmov<DPP_QUAD_XOR1>(x));
  x = fmaxf(x, dppmov<DPP_QUAD_XOR2>(x));
  x = fmaxf(x, dppmov<DPP_HALF_MIRROR>(x));
  x = fmaxf(x, dppmov<DPP_ROW_MIRROR>(x));
  return x;
}
__device__ __forceinline__ float row16_sum(float x) {
  x += dppmov<DPP_QUAD_XOR1>(x);
  x += dppmov<DPP_QUAD_XOR2>(x);
  x += dppmov<DPP_HALF_MIRROR>(x);
  x += dppmov<DPP_ROW_MIRROR>(x);
  return x;
}

// ---------------------------------------------------------------------------
// Kernel 1: fused Q/K/V projections + center-tap conv1d + K_new linear.
// One block (64 threads) per node. Writes zero-padded [N, FP] buffers.
// ---------------------------------------------------------------------------
__global__ void qkv_kernel(const float* __restrict__ x,
                           const float* __restrict__ Wq, const float* __restrict__ bq,
                           const float* __restrict__ Wk, const float* __restrict__ bk,
                           const float* __restrict__ Wv, const float* __restrict__ bv,
                           const float* __restrict__ W3, const float* __restrict__ b3,
                           const float* __restrict__ W5, const float* __restrict__ b5,
                           const float* __restrict__ Wl, const float* __restrict__ bl,
                           float* __restrict__ Qp, float* __restrict__ Vp,
                           float* __restrict__ Kn)
{
  const int n = blockIdx.x;
  const int t = threadIdx.x;
  __shared__ float xs[F], Ks[F], c3s[F], c5s[F];
  if (t < F) xs[t] = x[(size_t)n * F + t];
  __syncthreads();
  if (t < F) {
    float q = bq[t], k = bk[t], v = bv[t];
    for (int j = 0; j < F; ++j) {
      const float xv = xs[j];
      q += Wq[t * F + j] * xv;
      k += Wk[t * F + j] * xv;
      v += Wv[t * F + j] * xv;
    }
    Qp[(size_t)n * FP + t] = q;
    Vp[(size_t)n * FP + t] = v;
    Ks[t] = k;
  } else if (t < FP) {
    Qp[(size_t)n * FP + t] = 0.f;
    Vp[(size_t)n * FP + t] = 0.f;
    Kn[(size_t)n * FP + t] = 0.f;
  }
  __syncthreads();
  if (t < F) {
    float a = b3[t], b = b5[t];
    for (int j = 0; j < F; ++j) {
      a += W3[(t * F + j) * 3 + 1] * Ks[j];  // center tap of k=3 conv
      b += W5[(t * F + j) * 5 + 2] * Ks[j];  // center tap of k=5 conv
    }
    c3s[t] = a; c5s[t] = b;
  }
  __syncthreads();
  if (t < F) {
    float kn = bl[t];
    const float* wr = Wl + (size_t)t * (3 * F);
    for (int j = 0; j < F; ++j)
      kn += wr[j] * c3s[j] + wr[F + j] * c5s[j] + wr[2 * F + j] * Ks[j];
    Kn[(size_t)n * FP + t] = kn;
  }
}

// ---------------------------------------------------------------------------
// Kernel 2: flash attention, fp32 WMMA 16x16x4. One wave = 16 output rows,
// 32 columns (two 16x16 tiles) per inner iteration so the DPP softmax
// butterflies and accumulator rescales are amortized over 2x the columns.
//   S = K_new * Q^T / sqrt(F); P = online-softmax(S); acc += P * V
//   h = relu(acc / l + V_row)
// ---------------------------------------------------------------------------
__global__ __launch_bounds__(128) void attn_kernel(
    const float* __restrict__ Qp, const float* __restrict__ Kn,
    const float* __restrict__ Vp, float* __restrict__ h, int N)
{
  __shared__ float Pl[4][2][16 * 20];              // per-wave P tiles, bank-padded
  const int wave = threadIdx.x >> 5;
  const int lane = threadIdx.x & 31;
  const int rb   = blockIdx.x * 4 + wave;
  if (rb * 16 >= N) return;                        // wave-uniform exit
  const int r0   = rb * 16;
  const int m15  = lane & 15;
  const int half = lane >> 4;
  float* P0 = &Pl[wave][0][0];
  float* P1 = &Pl[wave][1][0];

  // A fragments: 16 rows of K_new, 9 chunks of K=4 (covers F=35, zero padded)
  v2f afr[9];
  {
    const float* kr = Kn + (size_t)(r0 + m15) * FP + 2 * half;
    #pragma unroll
    for (int c = 0; c < 9; ++c) afr[c] = *(const v2f*)(kr + c * 4);
  }

  v8f acc[3];
  float m[8], l[8];
  #pragma unroll
  for (int r = 0; r < 8; ++r) {
    m[r] = -3.0e38f; l[r] = 0.f;
    acc[0][r] = 0.f; acc[1][r] = 0.f; acc[2][r] = 0.f;
  }
  const float scale = 0.16903085f;                 // 1/sqrt(35)

  for (int i0 = 0; i0 < N; i0 += 32) {
    // ---- two S tiles: 18 fp32 WMMAs (independent chains for ILP) ----
    v8f s0 = {0.f, 0.f, 0.f, 0.f, 0.f, 0.f, 0.f, 0.f};
    v8f s1 = {0.f, 0.f, 0.f, 0.f, 0.f, 0.f, 0.f, 0.f};
    const float* qr0 = Qp + (size_t)(i0 + m15) * FP + 2 * half;
    const float* qr1 = qr0 + (size_t)16 * FP;
    #pragma unroll
    for (int c = 0; c < 9; ++c) {
      v2f b0 = *(const v2f*)(qr0 + c * 4);
      v2f b1 = *(const v2f*)(qr1 + c * 4);
      s0 = __builtin_amdgcn_wmma_f32_16x16x4_f32(false, afr[c], false, b0,
                                                 (short)0, s0, false, false);
      s1 = __builtin_amdgcn_wmma_f32_16x16x4_f32(false, afr[c], false, b1,
                                                 (short)0, s1, false, false);
    }
    // ---- online softmax per row over 32 columns (one butterfly pair) ----
    #pragma unroll
    for (int r = 0; r < 8; ++r) {
      const float v0 = s0[r] * scale;
      const float v1 = s1[r] * scale;
      const float t  = row16_max(fmaxf(v0, v1));
      const float mn = fmaxf(m[r], t);
      const float sc = __expf(m[r] - mn);
      m[r] = mn;
      const float p0 = __expf(v0 - mn);
      const float p1 = __expf(v1 - mn);
      s0[r] = p0; s1[r] = p1;
      const float ts = row16_sum(p0 + p1);
      l[r] = l[r] * sc + ts;
      acc[0][r] *= sc; acc[1][r] *= sc; acc[2][r] *= sc;
    }
    // ---- transpose P tiles (D layout) -> A layout via per-wave LDS ----
    #pragma unroll
    for (int r = 0; r < 8; ++r) {
      const int a = (r + 8 * half) * 20 + m15;
      P0[a] = s0[r];
      P1[a] = s1[r];
    }
    v2f pf0[4], pf1[4];
    #pragma unroll
    for (int c = 0; c < 4; ++c) {
      const int a = m15 * 20 + c * 4 + 2 * half;
      pf0[c] = *(const v2f*)(P0 + a);
      pf1[c] = *(const v2f*)(P1 + a);
    }
    // ---- acc += P * V : 3 f-tiles x 2 col-tiles x 4 K-chunks = 24 WMMAs ----
    #pragma unroll
    for (int ft = 0; ft < 3; ++ft) {
      const float* vc0 = Vp + (size_t)i0 * FP + ft * 16 + m15;
      const float* vc1 = vc0 + (size_t)16 * FP;
      #pragma unroll
      for (int c = 0; c < 4; ++c) {
        const int kr = c * 4 + 2 * half;
        v2f bf0, bf1;
        bf0.x = vc0[(size_t)kr * FP];
        bf0.y = vc0[(size_t)(kr + 1) * FP];
        bf1.x = vc1[(size_t)kr * FP];
        bf1.y = vc1[(size_t)(kr + 1) * FP];
        acc[ft] = __builtin_amdgcn_wmma_f32_16x16x4_f32(false, pf0[c], false, bf0,
                                                        (short)0, acc[ft],
                                                        false, false);
        acc[ft] = __builtin_amdgcn_wmma_f32_16x16x4_f32(false, pf1[c], false, bf1,
                                                        (short)0, acc[ft],
                                                        false, false);
      }
    }
  }
  // ---- epilogue: h = relu(attn@V + V) ----
  #pragma unroll
  for (int ft = 0; ft < 3; ++ft) {
    #pragma unroll
    for (int r = 0; r < 8; ++r) {
      const int row = r0 + r + 8 * half;
      const int f   = ft * 16 + m15;
      const float v = acc[ft][r] / l[r] + Vp[(size_t)row * FP + f];
      h[(size_t)row * FP + f] = fmaxf(v, 0.f);
    }
  }
}

// ---------------------------------------------------------------------------
// Kernel 3: zero helper (agg/deg/pool re-init every call for graph replay)
// ---------------------------------------------------------------------------
__global__ void zero_kernel(float* __restrict__ p, size_t count) {
  size_t i = (size_t)blockIdx.x * blockDim.x + threadIdx.x;
  if (i < count) p[i] = 0.f;
}

// ---------------------------------------------------------------------------
// Kernel 4: edge scatter (SAGE mean aggregation numerator + degree)
// ---------------------------------------------------------------------------
__global__ void edge_kernel(const int* __restrict__ ei, const float* __restrict__ h,
                            float* __restrict__ agg, float* __restrict__ deg, int E)
{
  int e = blockIdx.x * blockDim.x + threadIdx.x;
  if (e >= E) return;
  const int s = ei[e];
  const int d = ei[E + e];
  atomicAdd(&deg[d], 1.0f);
  const float* hr = h + (size_t)s * FP;
  float* ar = agg + (size_t)d * FP;
  for (int f = 0; f < F; ++f) atomicAdd(&ar[f], hr[f]);
}

// ---------------------------------------------------------------------------
// Kernel 5: SAGE linear: h2 = relu(lin_l(agg/deg) + lin_r(h))
// ---------------------------------------------------------------------------
__global__ void sage_kernel(const float* __restrict__ agg, const float* __restrict__ deg,
                            const float* __restrict__ h,
                            const float* __restrict__ Wll, const float* __restrict__ bll,
                            const float* __restrict__ Wlr,
                            float* __restrict__ h2, int N)
{
  int idx = blockIdx.x * blockDim.x + threadIdx.x;
  if (idx >= N * F) return;
  const int n = idx / F, f = idx - n * F;
  const float inv = 1.0f / fmaxf(deg[n], 1.0f);
  float accv = bll[f];
  const float* ar = agg + (size_t)n * FP;
  const float* hr = h + (size_t)n * FP;
  const float* wl = Wll + (size_t)f * F;
  const float* wr = Wlr + (size_t)f * F;
  for (int j = 0; j < F; ++j) accv += wl[j] * (ar[j] * inv) + wr[j] * hr[j];
  h2[(size_t)n * FP + f] = fmaxf(accv, 0.f);
}

// ---------------------------------------------------------------------------
// Kernel 6: global max pool. h2 >= 0 (post-ReLU) so uint-bit atomicMax == fmax
// ---------------------------------------------------------------------------
__global__ void pool_kernel(const float* __restrict__ h2, const int* __restrict__ batch,
                            float* __restrict__ g, int N)
{
  int idx = blockIdx.x * blockDim.x + threadIdx.x;
  if (idx >= N * F) return;
  const int n = idx / F, f = idx - n * F;
  const int b = batch[n];
  const unsigned bits = __float_as_uint(h2[(size_t)n * FP + f]);
  atomicMax((unsigned int*)&g[(size_t)b * F + f], bits);
}

// ---------------------------------------------------------------------------
// Kernels 7-9: MLP head
// ---------------------------------------------------------------------------
__global__ void mlp1_kernel(const float* __restrict__ g, const float* __restrict__ Wg1,
                            const float* __restrict__ bg1, float* __restrict__ g1, int B)
{
  int idx = blockIdx.x * blockDim.x + threadIdx.x;
  if (idx >= B * 1500) return;
  const int b = idx / 1500, j = idx - b * 1500;
  float acc = bg1[j];
  const float* gr = g + (size_t)b * F;
  const float* wr = Wg1 + (size_t)j * F;
  for (int k = 0; k < F; ++k) acc += wr[k] * gr[k];
  g1[idx] = fmaxf(acc, 0.f);
}

__global__ void mlp2_kernel(const float* __restrict__ g1, const float* __restrict__ Wg2,
                            const float* __restrict__ bg2, float* __restrict__ g2, int B)
{
  int idx = blockIdx.x * blockDim.x + threadIdx.x;
  if (idx >= B * 128) return;
  const int b = idx >> 7, j = idx & 127;
  float acc = bg2[j];
  const float* gr = g1 + (size_t)b * 1500;
  const float* wr = Wg2 + (size_t)j * 1500;
  for (int k = 0; k < 1500; ++k) acc += wr[k] * gr[k];
  g2[idx] = acc;
}

__global__ void out_kernel(const float* __restrict__ g2, const float* __restrict__ Wo,
                           const float* __restrict__ bo, float* __restrict__ out, int B)
{
  int b = blockIdx.x * blockDim.x + threadIdx.x;
  if (b >= B) return;
  float acc = bo[0];
  const float* gr = g2 + (size_t)b * 128;
  for (int k = 0; k < 128; ++k) acc += Wo[k] * gr[k];
  out[b] = acc;
}

// ---------------------------------------------------------------------------
extern "C" void kernel_launch(void* const* d_in, const int* in_sizes, int n_in,
                              void* d_out, int out_size, void* d_ws, size_t ws_size,
                              hipStream_t stream)
{
  const float* x     = (const float*)d_in[0];
  const int*   ei    = (const int*)d_in[1];
  const int*   batch = (const int*)d_in[2];
  const float* Wq = (const float*)d_in[3];  const float* bq = (const float*)d_in[4];
  const float* Wk = (const float*)d_in[5];  const float* bk = (const float*)d_in[6];
  const float* Wv = (const float*)d_in[7];  const float* bv = (const float*)d_in[8];
  const float* W3 = (const float*)d_in[9];  const float* b3 = (const float*)d_in[10];
  const float* W5 = (const float*)d_in[11]; const float* b5 = (const float*)d_in[12];
  const float* Wl = (const float*)d_in[13]; const float* bl = (const float*)d_in[14];
  const float* Wll= (const float*)d_in[15]; const float* bll= (const float*)d_in[16];
  const float* Wlr= (const float*)d_in[17];
  const float* Wg1= (const float*)d_in[18]; const float* bg1= (const float*)d_in[19];
  const float* Wg2= (const float*)d_in[20]; const float* bg2= (const float*)d_in[21];
  const float* Wo = (const float*)d_in[22]; const float* bo = (const float*)d_in[23];
  float* out = (float*)d_out;

  const int N = in_sizes[0] / F;
  const int E = in_sizes[1] / 2;
  const int B = out_size;

  // workspace layout (floats); agg reuses Qp, h2 reuses Kn
  float* ws  = (float*)d_ws;
  float* Qp  = ws;                               // [N,FP]  (later: agg)
  float* Vp  = ws + 1 * (size_t)N * FP;          // [N,FP]
  float* Kn  = ws + 2 * (size_t)N * FP;          // [N,FP]  (later: h2)
  float* h   = ws + 3 * (size_t)N * FP;          // [N,FP]
  float* deg = ws + 4 * (size_t)N * FP;          // [N]
  float* g   = deg + N;                          // [B,F]
  float* g1  = g + (size_t)B * F;                // [B,1500]
  float* g2  = g1 + (size_t)B * 1500;            // [B,128]

  qkv_kernel<<<N, 64, 0, stream>>>(x, Wq, bq, Wk, bk, Wv, bv, W3, b3, W5, b5,
                                   Wl, bl, Qp, Vp, Kn);

  const int rowblocks = N / 16;                  // N = 12000 -> 750
  attn_kernel<<<(rowblocks + 3) / 4, 128, 0, stream>>>(Qp, Kn, Vp, h, N);

  const size_t zc = (size_t)N * FP;
  zero_kernel<<<(int)((zc + 255) / 256), 256, 0, stream>>>(Qp, zc);       // agg
  zero_kernel<<<(N + 255) / 256, 256, 0, stream>>>(deg, (size_t)N);
  zero_kernel<<<(B * F + 255) / 256, 256, 0, stream>>>(g, (size_t)B * F);

  edge_kernel<<<(E + 255) / 256, 256, 0, stream>>>(ei, h, Qp, deg, E);
  sage_kernel<<<(N * F + 255) / 256, 256, 0, stream>>>(Qp, deg, h, Wll, bll,
                                                       Wlr, Kn, N);
  pool_kernel<<<(N * F + 255) / 256, 256, 0, stream>>>(Kn, batch, g, N);

  mlp1_kernel<<<(B * 1500 + 255) / 256, 256, 0, stream>>>(g, Wg1, bg1, g1, B);
  mlp2_kernel<<<(B * 128 + 255) / 256, 256, 0, stream>>>(g1, Wg2, bg2, g2, B);
  out_kernel<<<(B + 63) / 64, 64, 0, stream>>>(g2, Wo, bo, out, B);
}